// IntensityLoss_89764816486828
// MI455X (gfx1250) — compile-verified
//
#include <hip/hip_runtime.h>

// CDNA5 (gfx1250) nearest-neighbor intensity loss.
//
// score(n,m) = |t_m|^2 - 2 p_n . t_m   (same argmin as full squared distance)
//            = A_n . B_m  with A_n=(px,py,pz,1), B_m=(-2tx,-2ty,-2tz,|t|^2)
// -> K=4 homogeneous dot product == V_WMMA_F32_16X16X4_F32 per 16x16 tile.
//
// A-operand layout (16x4 f32, 2 VGPRs): lanes 0-15 hold (K=0,K=1) of row M=lane,
//   lanes 16-31 hold (K=2,K=3) of row M=lane-16.
// B-operand layout (4x16 f32, 2 VGPRs): lanes 0-15 hold (K=0,K=1) of col N=lane,
//   lanes 16-31 hold (K=2,K=3) of col N=lane-16.
// D layout (16x16 f32, 8 VGPRs): VGPR r, lanes 0-15 -> row r, col=lane;
//   lanes 16-31 -> row r+8, col=lane-16.

typedef float v2f __attribute__((ext_vector_type(2)));
typedef float v8f __attribute__((ext_vector_type(8)));

#define BLOCK_THREADS     256
#define WAVES_PER_BLOCK   (BLOCK_THREADS / 32)
#define TARGETS_PER_CHUNK 2048   // 2048 * 16B = 32KB LDS

__global__ __launch_bounds__(BLOCK_THREADS)
void nn_intensity_loss_kernel(const float* __restrict__ pred,
                              const float* __restrict__ target,
                              float* __restrict__ out,
                              int N, int M) {
    // Preprocessed B vectors: (-2x, -2y, -2z, |t|^2)
    __shared__ float4 sB[TARGETS_PER_CHUNK];

    const int lane = threadIdx.x & 31;
    const int wave = threadIdx.x >> 5;
    const int half = lane >> 4;   // 0: lanes 0-15 (K=0,1), 1: lanes 16-31 (K=2,3)
    const int sub  = lane & 15;

    const int rowBase = (blockIdx.x * WAVES_PER_BLOCK + wave) * 16;

    // Load this wave's pred tile once; build the A operand in registers.
    const float4 p = ((const float4*)pred)[rowBase + sub];
    v2f a;
    a.x = half ? p.z : p.x;
    a.y = half ? 1.0f : p.y;
    const float predw = p.w;   // lane j holds intensity of pred row (j & 15)

    // Running per-(vgpr,lane) min score + argmin index.
    float best[8];
    int   bidx[8];
#pragma unroll
    for (int r = 0; r < 8; ++r) { best[r] = 3.4e38f; bidx[r] = 0; }

    // Per-lane byte offset within a 16-point B tile: sub*16 + half*8
    // -> all 32 lanes hit distinct LDS banks on each ds_load_b64.
    const int laneOff = sub * 16 + half * 8;

    for (int chunk = 0; chunk < M; chunk += TARGETS_PER_CHUNK) {
        // Cooperative preprocess of 2048 targets into LDS.
        for (int t = threadIdx.x; t < TARGETS_PER_CHUNK; t += BLOCK_THREADS) {
            const float4 q = ((const float4*)target)[chunk + t];
            float4 b;
            b.x = -2.0f * q.x;
            b.y = -2.0f * q.y;
            b.z = -2.0f * q.z;
            b.w = q.x * q.x + q.y * q.y + q.z * q.z;
            sB[t] = b;
        }
        __syncthreads();

        const char* sBb = (const char*)sB;
        // 4-way unrolled: issue all 4 independent ds_load_b64 first so the
        // scheduler can wait on descending dscnt and overlap LDS latency with
        // WMMA + argmin-select work instead of stalling per tile.
        for (int tile = 0; tile < TARGETS_PER_CHUNK; tile += 64) {
            const v2f b0 = *(const v2f*)(sBb + (size_t)(tile +  0) * 16 + laneOff);
            const v2f b1 = *(const v2f*)(sBb + (size_t)(tile + 16) * 16 + laneOff);
            const v2f b2 = *(const v2f*)(sBb + (size_t)(tile + 32) * 16 + laneOff);
            const v2f b3 = *(const v2f*)(sBb + (size_t)(tile + 48) * 16 + laneOff);

            v8f c0 = {}, c1 = {}, c2 = {}, c3 = {};
            c0 = __builtin_amdgcn_wmma_f32_16x16x4_f32(false, a, false, b0,
                                                       (short)0, c0, false, false);
            c1 = __builtin_amdgcn_wmma_f32_16x16x4_f32(false, a, false, b1,
                                                       (short)0, c1, false, false);
            c2 = __builtin_amdgcn_wmma_f32_16x16x4_f32(false, a, false, b2,
                                                       (short)0, c2, false, false);
            c3 = __builtin_amdgcn_wmma_f32_16x16x4_f32(false, a, false, b3,
                                                       (short)0, c3, false, false);

            const int m0 = chunk + tile + sub;
#pragma unroll
            for (int r = 0; r < 8; ++r) {
                // Strict '<' keeps the earliest m on ties (m increases per
                // lane), matching jnp.argmin first-occurrence semantics.
                if (c0[r] < best[r]) { best[r] = c0[r]; bidx[r] = m0;      }
                if (c1[r] < best[r]) { best[r] = c1[r]; bidx[r] = m0 + 16; }
                if (c2[r] < best[r]) { best[r] = c2[r]; bidx[r] = m0 + 32; }
                if (c3[r] < best[r]) { best[r] = c3[r]; bidx[r] = m0 + 48; }
            }
        }
        __syncthreads();
    }

    // Finalize: butterfly-min within each 16-lane half (xor masks 1..8 never
    // cross the half boundary), tie-break toward the smaller target index.
    float lsum = 0.0f;
#pragma unroll
    for (int r = 0; r < 8; ++r) {
        float s = best[r];
        int   i = bidx[r];
#pragma unroll
        for (int off = 1; off < 16; off <<= 1) {
            const float s2 = __shfl_xor(s, off, 32);
            const int   i2 = __shfl_xor(i, off, 32);
            if (s2 < s || (s2 == s && i2 < i)) { s = s2; i = i2; }
        }
        // Row handled by this (vgpr, half): r for lanes 0-15, r+8 for lanes 16-31.
        const int row = r + half * 8;
        const float pint = __shfl(predw, row, 32);  // all lanes execute (src lane must be active)
        if (sub == 0) {
            const float tint = target[4 * i + 3];   // matched target intensity
            const float d = pint - tint;
            lsum += d * d;
        }
    }

    if (sub == 0) {
        // mean over N, LOSS_WEIGHT = 1.0
        atomicAdd(out, lsum / (float)N);
    }
}

extern "C" void kernel_launch(void* const* d_in, const int* in_sizes, int n_in,
                              void* d_out, int out_size, void* d_ws, size_t ws_size,
                              hipStream_t stream) {
    const float* pred   = (const float*)d_in[0];
    const float* target = (const float*)d_in[1];
    float* out = (float*)d_out;

    const int N = in_sizes[0] / 4;   // 16384 pred points
    const int M = in_sizes[1] / 4;   // 16384 target points

    // Zero the scalar accumulator (graph-capture-safe async memset).
    hipMemsetAsync(out, 0, sizeof(float), stream);

    const int blocks = N / (16 * WAVES_PER_BLOCK);  // 128 blocks, 8 waves each
    nn_intensity_loss_kernel<<<blocks, BLOCK_THREADS, 0, stream>>>(pred, target, out, N, M);
}